// RandomShift_69398081568864
// MI455X (gfx1250) — compile-verified
//
#include <hip/hip_runtime.h>
#include <stdint.h>

// Problem constants (must match the reference):
//   N=16, T=4096, F=256, PROP=0.3 -> TP = T + 2*floor(0.3*T) = 4096 + 2*1228 = 6552
#define N_  16
#define T_  4096
#define F_  256
#define TP_ 6552
#define R_  8          // rows per block; TP_ % R_ == 0 (6552 = 8 * 819)

typedef float v4f __attribute__((ext_vector_type(4)));

// One wave32 per block; each block produces R_ consecutive output rows
// out[n, t0..t0+R_-1, 0:256]. Valid rows move HBM/L2 -> LDS -> HBM through the
// CDNA5 async data-mover path (ASYNCcnt): all 2*R_ 128b async loads are issued
// back-to-back, one s_wait_asynccnt 0, then all async stores — deep MLP per
// wave and 8x amortization of the scalar prologue.
__global__ __launch_bounds__(32) void reflect_pad_gather(
    const float* __restrict__ in_,
    const int*   __restrict__ in_lens,
    const int*   __restrict__ pad,      // shape (2, N): pad[n], pad[N+n]
    float*       __restrict__ out)      // N*TP*F floats, then N int32 out_lens
{
    __shared__ __align__(16) float smem[R_][F_];   // 8 KB staging

    const int t0   = blockIdx.x * R_;
    const int n    = blockIdx.y;
    const int lane = threadIdx.x;                  // 0..31 (wave32)

    const int len = in_lens[n];
    const int p0  = pad[n];
    const int p1  = pad[N_ + n];
    const int out_len = len + p0 + p1;             // <= TP by construction

    // Second tuple output: out_lens, stored as int32 bit patterns after the
    // float payload.
    if (t0 == 0 && lane == 0) {
        ((int*)(out + (size_t)N_ * TP_ * F_))[n] = out_len;
    }

    const uint32_t lbase = (uint32_t)(uintptr_t)&smem[0][0] + (uint32_t)lane * 16u;

    bool valid[R_];

    // Phase 1: issue all async loads (valid rows) / zero-fill (invalid rows).
    #pragma unroll
    for (int r = 0; r < R_; ++r) {
        const int t = t0 + r;
        valid[r] = (t < out_len);
        float* dst = out + ((size_t)n * TP_ + t) * F_;

        if (valid[r]) {
            // Reflect-mode source index (matches reference clip semantics).
            int src;
            if (t < p0)            src = p0 - t;                // left reflect
            else if (t < p0 + len) src = t - p0;                // pass-through
            else                   src = 2 * len + p0 - t - 2;  // right reflect
            src = min(max(src, 0), T_ - 1);

            const char* gsrc = (const char*)(in_ + ((size_t)n * T_ + src) * F_)
                               + lane * 16;
            const uint32_t laddr = lbase + (uint32_t)r * (F_ * 4);

            // HBM -> LDS: two 128b async loads per lane cover this 1 KB row.
            // The instruction offset applies to both LDS and global sides.
            asm volatile("global_load_async_to_lds_b128 %0, %1, off"
                         :: "v"(laddr), "v"(gsrc) : "memory");
            asm volatile("global_load_async_to_lds_b128 %0, %1, off offset:512"
                         :: "v"(laddr), "v"(gsrc) : "memory");
        } else {
            // Invalid tail: reference zeroes it, and the harness poisons
            // d_out, so every element must be written. Non-temporal: this
            // data is never re-read; keep the input resident in L2.
            v4f  z  = {0.f, 0.f, 0.f, 0.f};
            v4f* d4 = (v4f*)dst;
            __builtin_nontemporal_store(z, d4 + lane);        // bytes [0,512)
            __builtin_nontemporal_store(z, d4 + lane + 32);   // bytes [512,1024)
        }
    }

    // Single wait for all outstanding async loads to land in LDS.
    asm volatile("s_wait_asynccnt 0" ::: "memory");

    // Phase 2: issue all async stores.
    #pragma unroll
    for (int r = 0; r < R_; ++r) {
        if (valid[r]) {
            char* gdst = (char*)(out + ((size_t)n * TP_ + (t0 + r)) * F_)
                         + lane * 16;
            const uint32_t laddr = lbase + (uint32_t)r * (F_ * 4);

            asm volatile("global_store_async_from_lds_b128 %0, %1, off"
                         :: "v"(gdst), "v"(laddr) : "memory");
            asm volatile("global_store_async_from_lds_b128 %0, %1, off offset:512"
                         :: "v"(gdst), "v"(laddr) : "memory");
        }
    }
    // S_ENDPGM performs an implicit wait-idle; no trailing wait needed, and
    // LDS is private to this block for its lifetime.
}

extern "C" void kernel_launch(void* const* d_in, const int* in_sizes, int n_in,
                              void* d_out, int out_size, void* d_ws, size_t ws_size,
                              hipStream_t stream) {
    (void)in_sizes; (void)n_in; (void)out_size; (void)d_ws; (void)ws_size;
    const float* in_     = (const float*)d_in[0];
    const int*   in_lens = (const int*)d_in[1];
    const int*   pad     = (const int*)d_in[2];
    float*       out     = (float*)d_out;

    dim3 grid(TP_ / R_, N_);
    reflect_pad_gather<<<grid, 32, 0, stream>>>(in_, in_lens, pad, out);
}